// DNC_47871705481292
// MI455X (gfx1250) — compile-verified
//
#include <hip/hip_runtime.h>
#include <hip/hip_bf16.h>

// ---------------------------------------------------------------------------
// DNC forward for MI455X (gfx1250, wave32).
// All GEMMs use V_WMMA_F32_16X16X4_F32 (full fp32 fidelity; workload is
// L2-resident and latency bound, so fp32 WMMA costs nothing vs bf16).
// B=16 maps exactly onto the WMMA M=16 tile dimension.
// ---------------------------------------------------------------------------

typedef __attribute__((ext_vector_type(2))) float v2f;
typedef __attribute__((ext_vector_type(8))) float v8f;

#define DNC_B   16
#define DNC_T   64
#define DNC_IN  256
#define DNC_H   512
#define DNC_M   256
#define DNC_W   64
#define DNC_R   4
#define DNC_IF  471
#define DNC_DELTA 1e-6f

__device__ __forceinline__ float sigm(float x) { return 1.0f / (1.0f + expf(-x)); }
__device__ __forceinline__ float softplusf(float x) {
  return x > 0.0f ? x + log1pf(expf(-x)) : log1pf(expf(x));
}

// ---------------------------------------------------------------------------
// zero-init for recurrent state
// ---------------------------------------------------------------------------
__global__ void zero_kernel(float* __restrict__ p, int n) {
  int i = blockIdx.x * blockDim.x + threadIdx.x;
  if (i < n) p[i] = 0.0f;
}

// ---------------------------------------------------------------------------
// Y(16 x O) = A1(16 x K1) @ W1[:, :K1]^T + A2(16 x K2) @ W2^T + bias(+bias2)
// W row-major (O x ldw); A rows strided by ldA.  One wave computes one 16x16
// output tile with V_WMMA_F32_16X16X4_F32; A staged in LDS (772-float stride
// breaks the 64-bank conflict).  8 waves / block.
// ---------------------------------------------------------------------------
#define A_STRIDE 772
__global__ __launch_bounds__(256) void gemm_dual_kernel(
    const float* __restrict__ A1, int ldA1, const float* __restrict__ W1, int ldw1, int K1,
    const float* __restrict__ A2, int ldA2, const float* __restrict__ W2, int ldw2, int K2,
    const float* __restrict__ bias, const float* __restrict__ bias2,
    float* __restrict__ Y, int ldy, int O, int nTiles)
{
  __shared__ float As[16 * A_STRIDE];
  const int t = threadIdx.x;
  const int Ktot = K1 + K2;
  for (int i = t; i < 16 * Ktot; i += 256) {
    int r = i / Ktot, k = i - r * Ktot;
    float v = (k < K1) ? A1[(long)r * ldA1 + k] : A2[(long)r * ldA2 + (k - K1)];
    As[r * A_STRIDE + k] = v;
  }
  __syncthreads();

  const int tile = blockIdx.x * 8 + (t >> 5);
  if (tile >= nTiles) return;          // uniform per wave; barrier already passed
  const int lane = t & 31;
  const int row  = lane & 15;          // A row (M) and B column (N)
  const int kh   = (lane >> 4) << 1;   // K pair select: 0 or 2
  const int o0   = tile * 16;
  int oc = o0 + row;  if (oc >= O) oc = O - 1;   // clamped (stores are guarded)

  v8f acc = {};
  const float* a1 = As + row * A_STRIDE;
  const float* w1 = W1 + (long)oc * ldw1;
  for (int k0 = 0; k0 < K1; k0 += 4) {
    v2f av = { a1[k0 + kh], a1[k0 + kh + 1] };
    v2f bv = { w1[k0 + kh], w1[k0 + kh + 1] };
    acc = __builtin_amdgcn_wmma_f32_16x16x4_f32(false, av, false, bv, (short)0, acc, false, false);
  }
  if (K2 > 0) {
    const float* a2 = As + row * A_STRIDE + K1;
    const float* w2 = W2 + (long)oc * ldw2;
    for (int k0 = 0; k0 < K2; k0 += 4) {
      v2f av = { a2[k0 + kh], a2[k0 + kh + 1] };
      v2f bv = { w2[k0 + kh], w2[k0 + kh + 1] };
      acc = __builtin_amdgcn_wmma_f32_16x16x4_f32(false, av, false, bv, (short)0, acc, false, false);
    }
  }
  const int n = o0 + row;
  if (n < O) {
    float bb = (bias ? bias[n] : 0.0f) + (bias2 ? bias2[n] : 0.0f);
    const int mofs = (lane >> 4) * 8;        // D layout: VGPR i -> M=i (+8 for hi lanes)
    #pragma unroll
    for (int i = 0; i < 8; ++i) {
      int m = i + mofs;
      Y[(long)m * ldy + n] = acc[i] + bb;
    }
  }
}

// ---------------------------------------------------------------------------
// LSTM pointwise: gates (16 x 2048) order i,f,g,o -> update h,c (16 x 512)
// ---------------------------------------------------------------------------
__global__ __launch_bounds__(256) void lstm_pw_kernel(
    const float* __restrict__ gates, float* __restrict__ h, float* __restrict__ c)
{
  int idx = blockIdx.x * 256 + threadIdx.x;      // 16*512
  int b = idx >> 9, j = idx & 511;
  const float* g = gates + b * 2048;
  float ig = sigm(g[j]);
  float fg = sigm(g[512 + j]);
  float gg = tanhf(g[1024 + j]);
  float og = sigm(g[1536 + j]);
  float cn = fg * c[idx] + ig * gg;
  c[idx] = cn;
  h[idx] = og * tanhf(cn);
}

// ---------------------------------------------------------------------------
// Interface + usage + allocation (exact stable argsort via rank) + write
// weighting + memory erase/write + precedence.  One block per batch item.
// ---------------------------------------------------------------------------
__global__ __launch_bounds__(256) void iface_kernel(
    const float* __restrict__ xib,          // 16 x 471
    float* __restrict__ mem,                // 16 x 256 x 64
    float* __restrict__ usage,              // 16 x 256
    float* __restrict__ ww,                 // 16 x 256 (old in, new out)
    const float* __restrict__ rw,           // 16 x 4 x 256 (previous)
    const float* __restrict__ prec,         // 16 x 256 (old)
    float* __restrict__ precn,              // 16 x 256 (new)
    float* __restrict__ rkh_g,              // 16 x 256 normalized read keys
    float* __restrict__ rs_g,               // 16 x 4
    float* __restrict__ rm_g)               // 16 x 12
{
  const int b = blockIdx.x;
  const int t = threadIdx.x;
  __shared__ float xi[DNC_IF];
  __shared__ float wk_s[64], ev_s[64], wv_s[64];
  __shared__ float fg_s[4], rm_s[12], nrm_s[4], sc_s[4];  // sc: ag, wg, ws, |wk|+d
  __shared__ float u_s[256], a2_s[256], srt_s[256], pe_s[256], wcw_s[256], ww_s[256];
  __shared__ float red[256];

  const float* xr = xib + b * DNC_IF;
  for (int i = t; i < DNC_IF; i += 256) xi[i] = xr[i];
  __syncthreads();

  // ---- parse interface vector ----
  if (t < 64) {
    wk_s[t] = tanhf(xi[260 + t]);
    ev_s[t] = sigm (xi[325 + t]);
    wv_s[t] = tanhf(xi[389 + t]);
  }
  if (t < 4) {
    fg_s[t] = sigm(xi[453 + t]);
    float s = 0.0f;
    for (int w = 0; w < 64; ++w) { float v = tanhf(xi[t * 64 + w]); s += v * v; }
    nrm_s[t] = sqrtf(s) + DNC_DELTA;
    rs_g[b * 4 + t] = softplusf(xi[256 + t]);
    // read-mode softmax (rows of 3)
    float a0 = xi[459 + t * 3], a1 = xi[460 + t * 3], a2 = xi[461 + t * 3];
    float mx = fmaxf(a0, fmaxf(a1, a2));
    float e0 = expf(a0 - mx), e1 = expf(a1 - mx), e2 = expf(a2 - mx);
    float s3 = e0 + e1 + e2;
    rm_s[t * 3 + 0] = e0 / s3; rm_s[t * 3 + 1] = e1 / s3; rm_s[t * 3 + 2] = e2 / s3;
  }
  if (t == 0) {
    sc_s[0] = sigm(xi[457]);          // alloc gate
    sc_s[1] = sigm(xi[458]);          // write gate
    sc_s[2] = softplusf(xi[324]);     // write strength
  }
  __syncthreads();
  if (t == 0) {
    float s = 0.0f;
    for (int w = 0; w < 64; ++w) s += wk_s[w] * wk_s[w];
    sc_s[3] = sqrtf(s) + DNC_DELTA;   // |wk| + delta
  }
  if (t < 12) rm_g[b * 12 + t] = rm_s[t];
  // normalized read keys to global
  {
    int r = t >> 6, w = t & 63;
    rkh_g[b * 256 + t] = tanhf(xi[r * 64 + w]) / nrm_s[r];
  }
  __syncthreads();

  // ---- usage update (uses OLD ww and OLD rw) ----
  float ww_old = ww[b * 256 + t];
  float u = usage[b * 256 + t];
  u = u + (1.0f - u) * ww_old;
  float psi = 1.0f;
  #pragma unroll
  for (int r = 0; r < 4; ++r)
    psi *= 1.0f - fg_s[r] * rw[b * 1024 + r * 256 + t];
  u *= psi;
  u_s[t] = u;
  usage[b * 256 + t] = u;

  // ---- write content weighting (OLD mem) ----
  float nm = 0.0f, dt = 0.0f;
  {
    const float* mr = mem + ((size_t)b * 256 + t) * 64;
    for (int w = 0; w < 64; ++w) { float mv = mr[w]; nm += mv * mv; dt += mv * wk_s[w]; }
  }
  float score = dt / ((sqrtf(nm) + DNC_DELTA) * sc_s[3]) * sc_s[2];
  red[t] = score; __syncthreads();
  for (int s = 128; s > 0; s >>= 1) { if (t < s) red[t] = fmaxf(red[t], red[t + s]); __syncthreads(); }
  float mx = red[0]; __syncthreads();
  float ex = expf(score - mx);
  red[t] = ex; __syncthreads();
  for (int s = 128; s > 0; s >>= 1) { if (t < s) red[t] += red[t + s]; __syncthreads(); }
  wcw_s[t] = ex / red[0];
  __syncthreads();

  // ---- allocation: stable-argsort semantics via rank ----
  float u2 = DNC_DELTA + (1.0f - DNC_DELTA) * u_s[t];
  a2_s[t] = u2;
  __syncthreads();
  int rank = 0;
  for (int j = 0; j < 256; ++j) {
    float vj = a2_s[j];
    rank += (vj < u2) || (vj == u2 && j < t);
  }
  srt_s[rank] = u2;
  __syncthreads();
  if (t == 0) {
    float run = 1.0f;
    for (int i = 0; i < 256; ++i) { pe_s[i] = run; run *= srt_s[i]; }
  }
  __syncthreads();
  float alloc = (1.0f - u2) * pe_s[rank];

  // ---- write weighting ----
  float ag = sc_s[0], wg = sc_s[1];
  float ww_new = wg * (ag * alloc + (1.0f - ag) * wcw_s[t]);
  ww_s[t] = ww_new;
  ww[b * 256 + t] = ww_new;
  red[t] = ww_new; __syncthreads();
  for (int s = 128; s > 0; s >>= 1) { if (t < s) red[t] += red[t + s]; __syncthreads(); }
  float sww = red[0];
  precn[b * 256 + t] = (1.0f - sww) * prec[b * 256 + t] + ww_new;
  __syncthreads();   // everyone done reading OLD mem before rewrite

  // ---- memory erase + write (coalesced) ----
  float* mb = mem + (size_t)b * 16384;
  for (int i = t; i < 16384; i += 256) {
    int m = i >> 6, w = i & 63;
    float wwm = ww_s[m];
    mb[i] = mb[i] * (1.0f - wwm * ev_s[w]) + wwm * wv_s[w];
  }
}

// ---------------------------------------------------------------------------
// Temporal link matrix update (reads OLD prec), diagonal zeroed.
// grid (8 row-blocks, 16 batch); thread = column n, 32 rows each.
// ---------------------------------------------------------------------------
__global__ __launch_bounds__(256) void link_kernel(
    float* __restrict__ link, const float* __restrict__ ww, const float* __restrict__ prec)
{
  const int b  = blockIdx.y;
  const int rb = blockIdx.x;
  const int t  = threadIdx.x;
  __shared__ float ww_s[256], pr_s[256];
  ww_s[t] = ww[b * 256 + t];
  pr_s[t] = prec[b * 256 + t];
  __syncthreads();
  float wn = ww_s[t], pn = pr_s[t];
  float* Lb = link + (size_t)b * 65536;
  for (int m = rb * 32; m < rb * 32 + 32; ++m) {
    float L = Lb[m * 256 + t];
    L = (1.0f - ww_s[m] - wn) * L + ww_s[m] * pn;
    if (m == t) L = 0.0f;
    Lb[m * 256 + t] = L;
  }
}

// ---------------------------------------------------------------------------
// fwd[r,m] = sum_n L[m,n]*rw[r,n]   bwd[r,n] = sum_m rw[r,m]*L[m,n]
// thread t owns row t (fwd) and column t (bwd).
// ---------------------------------------------------------------------------
__global__ __launch_bounds__(256) void dirw_kernel(
    const float* __restrict__ link, const float* __restrict__ rw,
    float* __restrict__ fwd, float* __restrict__ bwd)
{
  const int b = blockIdx.x;
  const int t = threadIdx.x;
  __shared__ float rw_s[1024];
  for (int i = t; i < 1024; i += 256) rw_s[i] = rw[b * 1024 + i];
  __syncthreads();
  const float* Lb = link + (size_t)b * 65536;
  float f0 = 0, f1 = 0, f2 = 0, f3 = 0, g0 = 0, g1 = 0, g2 = 0, g3 = 0;
  for (int n = 0; n < 256; ++n) {
    float lr = Lb[t * 256 + n];
    f0 += lr * rw_s[n];       f1 += lr * rw_s[256 + n];
    f2 += lr * rw_s[512 + n]; f3 += lr * rw_s[768 + n];
    float lc = Lb[n * 256 + t];
    g0 += rw_s[n] * lc;       g1 += rw_s[256 + n] * lc;
    g2 += rw_s[512 + n] * lc; g3 += rw_s[768 + n] * lc;
  }
  fwd[b * 1024 +       t] = f0;  fwd[b * 1024 + 256 + t] = f1;
  fwd[b * 1024 + 512 + t] = f2;  fwd[b * 1024 + 768 + t] = f3;
  bwd[b * 1024 +       t] = g0;  bwd[b * 1024 + 256 + t] = g1;
  bwd[b * 1024 + 512 + t] = g2;  bwd[b * 1024 + 768 + t] = g3;
}

// ---------------------------------------------------------------------------
// Read content weights (NEW mem), mode mix -> new rw, read vectors; also
// commits prec_new -> prec.
// ---------------------------------------------------------------------------
__global__ __launch_bounds__(256) void read_kernel(
    const float* __restrict__ mem, const float* __restrict__ rkh,
    const float* __restrict__ rs, const float* __restrict__ rm,
    const float* __restrict__ fwd, const float* __restrict__ bwd,
    float* __restrict__ rw, float* __restrict__ rv,
    float* __restrict__ prec, const float* __restrict__ precn)
{
  const int b = blockIdx.x;
  const int t = threadIdx.x;
  __shared__ float rk_s[256];
  __shared__ float rs_s[4], rm_s[12];
  __shared__ float cw_s[1024], rwn_s[1024];
  __shared__ float red[256];
  rk_s[t] = rkh[b * 256 + t];
  if (t < 4)  rs_s[t] = rs[b * 4 + t];
  if (t < 12) rm_s[t] = rm[b * 12 + t];
  __syncthreads();

  const float* mr = mem + ((size_t)b * 256 + t) * 64;
  float nm = 0, d0 = 0, d1 = 0, d2 = 0, d3 = 0;
  for (int w = 0; w < 64; ++w) {
    float mv = mr[w];
    nm += mv * mv;
    d0 += mv * rk_s[w];       d1 += mv * rk_s[64 + w];
    d2 += mv * rk_s[128 + w]; d3 += mv * rk_s[192 + w];
  }
  float inv = 1.0f / (sqrtf(nm) + DNC_DELTA);
  float scr[4] = { d0 * inv * rs_s[0], d1 * inv * rs_s[1],
                   d2 * inv * rs_s[2], d3 * inv * rs_s[3] };
  for (int r = 0; r < 4; ++r) {
    red[t] = scr[r]; __syncthreads();
    for (int s = 128; s > 0; s >>= 1) { if (t < s) red[t] = fmaxf(red[t], red[t + s]); __syncthreads(); }
    float mx = red[0]; __syncthreads();
    float e = expf(scr[r] - mx);
    red[t] = e; __syncthreads();
    for (int s = 128; s > 0; s >>= 1) { if (t < s) red[t] += red[t + s]; __syncthreads(); }
    cw_s[r * 256 + t] = e / red[0];
    __syncthreads();
  }
  #pragma unroll
  for (int r = 0; r < 4; ++r) {
    float v = rm_s[r * 3 + 0] * bwd[b * 1024 + r * 256 + t]
            + rm_s[r * 3 + 1] * fwd[b * 1024 + r * 256 + t]
            + rm_s[r * 3 + 2] * cw_s[r * 256 + t];
    rwn_s[r * 256 + t] = v;
    rw[b * 1024 + r * 256 + t] = v;
  }
  prec[b * 256 + t] = precn[b * 256 + t];
  __syncthreads();

  // read vectors: thread -> (r, w)
  int r = t >> 6, w = t & 63;
  float acc = 0.0f;
  const float* mb = mem + (size_t)b * 16384;
  for (int m = 0; m < 256; ++m) acc += rwn_s[r * 256 + m] * mb[m * 64 + w];
  rv[b * 256 + t] = acc;
}

// ---------------------------------------------------------------------------
// Host-side launch sequence (graph-capture safe: stream-only, no sync/alloc)
// ---------------------------------------------------------------------------
extern "C" void kernel_launch(void* const* d_in, const int* in_sizes, int n_in,
                              void* d_out, int out_size, void* d_ws, size_t ws_size,
                              hipStream_t stream)
{
  (void)in_sizes; (void)n_in; (void)out_size; (void)ws_size;
  const float* x    = (const float*)d_in[0];
  const float* Wih0 = (const float*)d_in[1];
  const float* bih0 = (const float*)d_in[2];
  const float* Whh0 = (const float*)d_in[3];
  const float* bhh0 = (const float*)d_in[4];
  const float* Wih1 = (const float*)d_in[5];
  const float* bih1 = (const float*)d_in[6];
  const float* Whh1 = (const float*)d_in[7];
  const float* bhh1 = (const float*)d_in[8];
  const float* Wout = (const float*)d_in[9];
  const float* bout = (const float*)d_in[10];
  const float* Wif  = (const float*)d_in[11];
  const float* bif  = (const float*)d_in[12];
  const float* Wmem = (const float*)d_in[13];
  const float* bmem = (const float*)d_in[14];
  float* out = (float*)d_out;
  float* ws  = (float*)d_ws;

  // --- recurrent state (zeroed every call; contiguous) ---
  float* h0    = ws;                   // 16*512
  float* c0    = h0    + 8192;
  float* h1    = c0    + 8192;
  float* c1    = h1    + 8192;
  float* mem   = c1    + 8192;         // 16*256*64
  float* link  = mem   + 262144;       // 16*256*256
  float* prec  = link  + 1048576;      // 16*256
  float* rw    = prec  + 4096;         // 16*4*256
  float* wwb   = rw    + 16384;        // 16*256
  float* usage = wwb   + 4096;         // 16*256
  int stateN   = (int)((usage + 4096) - ws);
  // --- per-step scratch ---
  float* gates = usage + 4096;         // 16*2048
  float* outb  = gates + 32768;        // 16*512
  float* xib   = outb  + 8192;         // 16*471
  float* precn = xib   + 16 * DNC_IF;  // 16*256
  float* fwd   = precn + 4096;         // 16*4*256
  float* bwd   = fwd   + 16384;        // 16*4*256
  float* rkh   = bwd   + 16384;        // 16*256
  float* rsb   = rkh   + 4096;         // 16*4
  float* rmb   = rsb   + 64;           // 16*12
  float* rvb   = rmb   + 192;          // 16*256

  zero_kernel<<<(stateN + 255) / 256, 256, 0, stream>>>(ws, stateN);

  for (int t = 0; t < DNC_T; ++t) {
    // LSTM0 gates: x_t part uses only first 256 cols of Wih0 (last_read == 0)
    gemm_dual_kernel<<<16, 256, 0, stream>>>(
        x + t * DNC_IN, DNC_T * DNC_IN, Wih0, 512, 256,
        h0, 512, Whh0, 512, 512,
        bih0, bhh0, gates, 2048, 2048, 128);
    lstm_pw_kernel<<<32, 256, 0, stream>>>(gates, h0, c0);

    gemm_dual_kernel<<<16, 256, 0, stream>>>(
        h0, 512, Wih1, 512, 512,
        h1, 512, Whh1, 512, 512,
        bih1, bhh1, gates, 2048, 2048, 128);
    lstm_pw_kernel<<<32, 256, 0, stream>>>(gates, h1, c1);

    gemm_dual_kernel<<<4, 256, 0, stream>>>(
        h1, 512, Wout, 512, 512,
        nullptr, 0, nullptr, 0, 0,
        bout, nullptr, outb, 512, 512, 32);
    gemm_dual_kernel<<<4, 256, 0, stream>>>(
        h1, 512, Wif, 512, 512,
        nullptr, 0, nullptr, 0, 0,
        bif, nullptr, xib, DNC_IF, DNC_IF, 30);

    iface_kernel<<<16, 256, 0, stream>>>(xib, mem, usage, wwb, rw, prec, precn,
                                         rkh, rsb, rmb);
    link_kernel<<<dim3(8, 16), 256, 0, stream>>>(link, wwb, prec);
    dirw_kernel<<<16, 256, 0, stream>>>(link, rw, fwd, bwd);
    read_kernel<<<16, 256, 0, stream>>>(mem, rkh, rsb, rmb, fwd, bwd,
                                        rw, rvb, prec, precn);

    // y = [out, read_vecs] @ Wmem.T + bmem  -> d_out[:, t, :]
    gemm_dual_kernel<<<2, 256, 0, stream>>>(
        outb, 512, Wmem, 768, 512,
        rvb, 256, Wmem + 512, 768, 256,
        bmem, nullptr, out + t * DNC_IN, DNC_T * DNC_IN, DNC_IN, 16);
  }
}